// SinkhornNet_50182397886973
// MI455X (gfx1250) — compile-verified
//
#include <hip/hip_runtime.h>
#include <hip/hip_bf16.h>

#define BATCH    64
#define NR_MAX   1024
#define MC       1024
#define MAX_IT   10
#define EPS_F    1e-4f

typedef __attribute__((ext_vector_type(2))) float v2f;
typedef __attribute__((ext_vector_type(8))) float v8f;

// One workgroup per batch element. u/v scale vectors live in LDS; the 4MB
// slab stays L2-resident across all 10 matvec passes + the final pass.
__global__ __launch_bounds__(1024, 1) void sinkhorn_fused(
    const float* __restrict__ s,
    const int*   __restrict__ nrows,
    const int*   __restrict__ ncols,
    float*       __restrict__ out)
{
    __shared__ float u_lds[NR_MAX];
    __shared__ float v_lds[MC];

    const int b    = blockIdx.x;
    const int tid  = threadIdx.x;
    const int lane = tid & 31;
    const int wave = tid >> 5;

    const int nr = nrows[b];
    const int nc = ncols[b];

    const float* __restrict__ sb = s   + (size_t)b * NR_MAX * MC;
    float*       __restrict__ ob = out + (size_t)b * NR_MAX * MC;

    // u = 1 inside valid rows, 0 outside; v = 0. Zero-padding of u/v makes
    // every inner loop mask-free and makes the final pass zero-fill for free.
    u_lds[tid] = (tid < nr) ? 1.0f : 0.0f;
    v_lds[tid] = 0.0f;
    __syncthreads();

    const int nc4 = (nc + 3) & ~3;   // row-phase K-loop bound (v==0 on tail)

    for (int it = 0; it < MAX_IT; ++it) {
        if ((it & 1) == 0) {
            // ---- column phase: v[j] = 1 / sum_{i<nr} (s+eps)*u[i] ----
            // thread j walks its column; loads are 4KB coalesced bursts.
            const float* colp = sb + tid;
            float sum = 0.0f;
            for (int i0 = 0; i0 < nr; i0 += 32) {
                int ipf = i0 + 96; if (ipf > NR_MAX - 1) ipf = NR_MAX - 1;
                __builtin_prefetch(colp + (size_t)ipf * MC, 0, 1); // global_prefetch_b8
                int iend = i0 + 32; if (iend > nr) iend = nr;
                #pragma unroll 8
                for (int i = i0; i < iend; ++i)
                    sum = fmaf(colp[(size_t)i * MC] + EPS_F, u_lds[i], sum);
            }
            v_lds[tid] = (tid < nc) ? (1.0f / sum) : 0.0f;
            __syncthreads();
        } else {
            // ---- row phase via V_WMMA_F32_16X16X4_F32 ----
            // D[m,n] = sum_k Aop[m,k]*Bop[k,n] with
            //   Aop[m,k] = v[jb+k]            (replicated over m)
            //   Bop[k,n] = s[rowbase+n][jb+k] + eps
            // => every column n of D = row-sum chunk for row rowbase+n,
            //    identical across m; accumulate chunks in C.
            // 4x unrolled K-loop: 4 global + 4 LDS loads in flight per wait;
            // two accumulators break the WMMA C-chain dependency.
            for (int rg = wave; rg < NR_MAX / 16; rg += 32) {
                const int rowbase = rg * 16;
                const int srow    = rowbase + (lane & 15);
                if (rowbase >= nr) {                 // wave-uniform skip
                    if (lane < 16) u_lds[srow] = 0.0f;
                    continue;
                }
                // A 16x4 layout: lanes 0-15 hold K={0,1}, lanes 16-31 K={2,3}
                const int koff = (lane >> 4) << 1;   // 0 or 2
                const float* browp = sb + (size_t)srow * MC + koff;
                v8f acc0 = {};
                v8f acc1 = {};
                int jb = 0;
                for (; jb + 16 <= nc4; jb += 16) {
                    v2f a0 = *(const v2f*)&v_lds[jb +  0 + koff];
                    v2f a1 = *(const v2f*)&v_lds[jb +  4 + koff];
                    v2f a2 = *(const v2f*)&v_lds[jb +  8 + koff];
                    v2f a3 = *(const v2f*)&v_lds[jb + 12 + koff];
                    v2f b0 = *(const v2f*)(browp + jb +  0);
                    v2f b1 = *(const v2f*)(browp + jb +  4);
                    v2f b2 = *(const v2f*)(browp + jb +  8);
                    v2f b3 = *(const v2f*)(browp + jb + 12);
                    b0.x += EPS_F; b0.y += EPS_F;
                    b1.x += EPS_F; b1.y += EPS_F;
                    b2.x += EPS_F; b2.y += EPS_F;
                    b3.x += EPS_F; b3.y += EPS_F;
                    acc0 = __builtin_amdgcn_wmma_f32_16x16x4_f32(
                        false, a0, false, b0, (short)0, acc0, false, false);
                    acc1 = __builtin_amdgcn_wmma_f32_16x16x4_f32(
                        false, a1, false, b1, (short)0, acc1, false, false);
                    acc0 = __builtin_amdgcn_wmma_f32_16x16x4_f32(
                        false, a2, false, b2, (short)0, acc0, false, false);
                    acc1 = __builtin_amdgcn_wmma_f32_16x16x4_f32(
                        false, a3, false, b3, (short)0, acc1, false, false);
                }
                for (; jb < nc4; jb += 4) {          // tail: <=3 chunks
                    v2f a  = *(const v2f*)&v_lds[jb + koff];
                    v2f bv = *(const v2f*)(browp + jb);
                    bv.x += EPS_F;
                    bv.y += EPS_F;
                    acc0 = __builtin_amdgcn_wmma_f32_16x16x4_f32(
                        false, a, false, bv, (short)0, acc0, false, false);
                }
                if (lane < 16) {                     // D constant over M: take acc[0]
                    float rs = acc0[0] + acc1[0];
                    u_lds[srow] = (srow < nr) ? (1.0f / rs) : 0.0f;
                }
            }
            __syncthreads();
        }
    }

    // ---- final: out = (s+eps) * u[i] * v[j]; zero-padded u/v mask for free ----
    const float vj = v_lds[tid];
    const float* colp = sb + tid;
    float* ocol = ob + tid;
    int i = 0;
    #pragma unroll 4
    for (; i < nr; ++i)
        ocol[(size_t)i * MC] = (colp[(size_t)i * MC] + EPS_F) * u_lds[i] * vj;
    for (; i < NR_MAX; ++i)
        ocol[(size_t)i * MC] = 0.0f;
}

extern "C" void kernel_launch(void* const* d_in, const int* in_sizes, int n_in,
                              void* d_out, int out_size, void* d_ws, size_t ws_size,
                              hipStream_t stream) {
    (void)in_sizes; (void)n_in; (void)out_size; (void)d_ws; (void)ws_size;
    const float* s     = (const float*)d_in[0];
    const int*   nrows = (const int*)d_in[1];
    const int*   ncols = (const int*)d_in[2];
    float*       out   = (float*)d_out;
    sinkhorn_fused<<<dim3(BATCH), dim3(1024), 0, stream>>>(s, nrows, ncols, out);
}